// OuterProduct_33904471835623
// MI455X (gfx1250) — compile-verified
//
#include <hip/hip_runtime.h>
#include <hip/hip_bf16.h>
#include <stdint.h>

// ---------------------------------------------------------------------------
// out[b,p,k] = sum_{a,c} x[b,ii(p),a] * x[b,jj(p),c] * W[p,k,a*64+c] + bias[p,k]
// P=120 GEMMs: M=1024(batch) x N=64(k) x K=4096(ac), A = on-the-fly outer product.
// fp32 WMMA (V_WMMA_F32_16X16X4_F32), W staged via TDM (tensor_load_to_lds)
// double-buffered in LDS with TDM row padding for bank-conflict-free B reads.
// ---------------------------------------------------------------------------

typedef __attribute__((ext_vector_type(2))) float v2f;
typedef __attribute__((ext_vector_type(8))) float v8f;
typedef __attribute__((ext_vector_type(4))) unsigned int v4u;
typedef __attribute__((ext_vector_type(8))) int v8i;
typedef __attribute__((ext_vector_type(4))) int v4i;

#define BB 1024
#define FF 16
#define DD 64
#define PP 120
#define KTOT (DD * DD)        // 4096
#define KC 64                 // K elements per LDS chunk (== one 'a' slice)
#define NCHUNK (KTOT / KC)    // 64
#define ROWS_PER_WAVE 16
#define WAVES 8
#define ROWS_PER_BLOCK (ROWS_PER_WAVE * WAVES)  // 128
#define LDS_STRIDE 68         // 64 data dwords + 4 pad dwords (TDM pad) -> conflict-free

// The TDM writes LDS behind the compiler's back (descriptor carries only an
// integer address). Passing the LDS pointer as an asm operand *captures* it,
// so with the memory clobber the compiler must assume this asm wrote `wt`
// and cannot fold the subsequent ds_load's to undef.
__device__ __forceinline__ void lds_written_fence(void* lds_ptr) {
  asm volatile("" : "+v"(lds_ptr) :: "memory");
}

// Issue one TDM load of a 64(n-rows) x 64(k) fp32 tile of W_p into LDS.
// Tensor: dim0 = 4096 (ac, contiguous), dim1 = 64 (rows n), stride0 = 4096.
// Tile:   tile_dim0 = 64, tile_dim1 = 64. Padding: 4 dwords per 64 dwords.
__device__ __forceinline__ void tdm_load_w_chunk(const float* gptr, unsigned lds_byte_addr)
{
  unsigned long long ga = (unsigned long long)(uintptr_t)gptr;
  v4u g0;
  g0.x = 1u;                                   // count=1, user descriptor
  g0.y = lds_byte_addr;                        // lds_addr (bytes)
  g0.z = (unsigned)(ga & 0xFFFFFFFFu);         // global_addr[31:0]
  g0.w = (unsigned)((ga >> 32) & 0x1FFFFFFu)   // global_addr[56:32]
         | (2u << 30);                         // type = 2 ("image")

  v8i g1;
  // data_size=4B(2), pad_enable=1, pad_interval=64 dwords(code 5), pad_amount=4 dwords(code 3)
  g1[0] = (int)((2u << 16) | (1u << 20) | (5u << 22) | (3u << 25));
  g1[1] = (int)((KTOT & 0xFFFFu) << 16);                   // tensor_dim0 lo16 (4096)
  g1[2] = (int)(((KTOT >> 16) & 0xFFFFu) | (DD << 16));    // tensor_dim0 hi | tensor_dim1=64 lo
  g1[3] = (int)(0u | (KC << 16));                          // tensor_dim1 hi | tile_dim0=64
  g1[4] = (int)(DD);                                       // tile_dim1=64, tile_dim2=0
  g1[5] = (int)KTOT;                                       // tensor_dim0_stride lo32 = 4096
  g1[6] = 0;                                               // stride0 hi | stride1 lo (unused, 2D)
  g1[7] = 0;
  v4i z4 = {0, 0, 0, 0};                                   // groups 2/3 unused (2D tensor)
  v8i z8 = {0, 0, 0, 0, 0, 0, 0, 0};
  __builtin_amdgcn_tensor_load_to_lds(g0, g1, z4, z4, z8, 0);
}

__global__ __launch_bounds__(256)
void fused_pair_outer_gemm(const float* __restrict__ x,
                           const float* __restrict__ W,
                           const float* __restrict__ bias,
                           float* __restrict__ out)
{
  __shared__ float wt[2][KC * LDS_STRIDE];

  // Low 32 bits of the generic pointer == LDS byte offset (aperture scheme:
  // the low dword is the LDS offset); used as the TDM descriptor's lds_addr.
  const unsigned lds_base = (unsigned)(uintptr_t)(void*)&wt[0][0];

  const int p    = blockIdx.y;
  const int bt   = blockIdx.x;
  const int tid  = threadIdx.x;
  const int lane = tid & 31;
  const int wid  = tid >> 5;
  const int m    = lane & 15;   // row-in-tile for A (batch), col-in-tile for B (k_out)
  const int half = lane >> 4;   // selects K slices {0,1} vs {2,3} in A/B operand layout

  // triu_indices(16, k=1) -> (ii, jj) for pair p (uniform scalar loop)
  int ii = 0;
  {
    int pp = p, cnt = FF - 1;
    while (pp >= cnt) { pp -= cnt; ++ii; --cnt; }
    const int jj_ = ii + 1 + pp;

    const int row = bt * ROWS_PER_BLOCK + wid * ROWS_PER_WAVE + m;  // batch row this lane feeds
    const float* xrow = x + (size_t)row * (FF * DD);
    const float* xi_p = xrow + ii * DD;
    const float* xj_p = xrow + jj_ * DD;

    // Preload the 16 xj pairs this lane needs: c = 4*s + 2*half (+1)
    v2f xj_pair[16];
#pragma unroll
    for (int s = 0; s < 16; ++s)
      xj_pair[s] = *(const v2f*)(xj_p + 4 * s + 2 * half);

    v8f acc[4] = {};  // N = 64 -> 4 tiles of 16

    const float* wp = W + (size_t)p * DD * KTOT;

    // Prologue: stage chunk 0
    if (wid == 0) {
      tdm_load_w_chunk(wp, lds_base);
      __builtin_amdgcn_s_wait_tensorcnt(0);
    }
    __syncthreads();
    lds_written_fence(&wt[0][0]);

    for (int ch = 0; ch < NCHUNK; ++ch) {
      const int cur = ch & 1;
      if (wid == 0 && (ch + 1) < NCHUNK) {
        tdm_load_w_chunk(wp + (size_t)(ch + 1) * KC,
                         lds_base + (unsigned)((cur ^ 1) * (KC * LDS_STRIDE * 4)));
      }

      const float xiv = xi_p[ch];          // a == chunk index
      const float* wtc = &wt[cur][0];

#pragma unroll
      for (int s = 0; s < 16; ++s) {
        const int c0 = 4 * s;
        v2f av;                             // A tile: lane supplies K = c0+2*half, +1
        av.x = xiv * xj_pair[s].x;
        av.y = xiv * xj_pair[s].y;
#pragma unroll
        for (int nt = 0; nt < 4; ++nt) {
          const int n = nt * 16 + m;        // B tile: lane supplies column n, rows 2*half,+1
          v2f bv = *(const v2f*)(wtc + n * LDS_STRIDE + c0 + 2 * half);
          acc[nt] = __builtin_amdgcn_wmma_f32_16x16x4_f32(
              false, av, false, bv, (short)0, acc[nt], false, false);
        }
      }

      if (wid == 0 && (ch + 1) < NCHUNK) __builtin_amdgcn_s_wait_tensorcnt(0);
      __syncthreads();
      lds_written_fence(&wt[cur ^ 1][0]);   // next buffer now valid
    }

    // Epilogue: bias + store. D layout: VGPR v -> M = v + 8*half, lane&15 -> N.
#pragma unroll
    for (int nt = 0; nt < 4; ++nt) {
      const int n = nt * 16 + m;
      const float bvv = bias[p * DD + n];
#pragma unroll
      for (int v = 0; v < 8; ++v) {
        const int orow = bt * ROWS_PER_BLOCK + wid * ROWS_PER_WAVE + v + 8 * half;
        out[(size_t)orow * (PP * DD) + (size_t)p * DD + n] = acc[nt][v] + bvv;
      }
    }
  }
}

extern "C" void kernel_launch(void* const* d_in, const int* in_sizes, int n_in,
                              void* d_out, int out_size, void* d_ws, size_t ws_size,
                              hipStream_t stream) {
  (void)in_sizes; (void)n_in; (void)d_ws; (void)ws_size; (void)out_size;
  const float* x    = (const float*)d_in[0];
  const float* W    = (const float*)d_in[1];
  const float* bias = (const float*)d_in[2];
  float* out        = (float*)d_out;

  dim3 grid(BB / ROWS_PER_BLOCK, PP);  // 8 x 120
  dim3 block(256);
  fused_pair_outer_gemm<<<grid, block, 0, stream>>>(x, W, bias, out);
}